// ClassWeightedHierarchicalGaussianMixture_62491774157086
// MI455X (gfx1250) — compile-verified
//
#include <hip/hip_runtime.h>
#include <math.h>

// Problem dims (fixed by reference setup_inputs)
#define B_DIM 2048
#define C_DIM 100
#define O_DIM 16
#define M_DIM 32
#define D_DIM 256
#define DG    8                 // d-groups for the main kernel
#define ND    (D_DIM / DG)      // 32 d's per wave
#define NBT   (B_DIM / 16)      // 128 b-tiles
#define NCT   ((C_DIM + 15)/16) // 7 c-tiles (padded to 112)

#define LOG2PI 1.8378770664093453f
#define LN2F   0.6931471805599453f

typedef float v2f __attribute__((ext_vector_type(2)));
typedef float v8f __attribute__((ext_vector_type(8)));

// Fast log: args are bounded away from denorms/inf in this problem, so the raw
// hardware v_log_f32 (log2) * ln2 is sufficient (2 instrs vs ~12 for __logf).
__device__ __forceinline__ float fast_log(float x) {
#if __has_builtin(__builtin_amdgcn_logf)
    return __builtin_amdgcn_logf(x) * LN2F;
#else
    return __logf(x);
#endif
}

// f32 WMMA 16x16x4 operand index map: for VGPR index i = 2k+j (k = K-step),
// the K element held by `lane` is 4k + j + (lane>=16 ? 2 : 0).
__device__ __forceinline__ int k_of(int i, int lane) {
    return 4 * (i >> 1) + (i & 1) + ((lane & 16) ? 2 : 0);
}

// ---------------------------------------------------------------------------
// P1: per-(m,d) quadratic coefficients of the Gaussian log-pdf, shifted by the
// per-d uniform logsumexp bound U[d] = max_m(-log_std) - 0.5*log(2pi).
// base[b,m,d] = A*z^2 + B*z + C'  with  C' = C - U[d]  (so base - U <= 0).
__global__ void prep_coefs(const float* __restrict__ means,
                           const float* __restrict__ logstds,
                           float* __restrict__ cA, float* __restrict__ cB,
                           float* __restrict__ cC, float* __restrict__ U) {
    int d = blockIdx.x * blockDim.x + threadIdx.x;
    if (d >= D_DIM) return;
    float umax = -INFINITY;
    for (int m = 0; m < M_DIM; ++m)
        umax = fmaxf(umax, -logstds[m * D_DIM + d]);
    float u = umax - 0.5f * LOG2PI;
    U[d] = u;
    for (int m = 0; m < M_DIM; ++m) {
        float ls = logstds[m * D_DIM + d];
        float mu = means[m * D_DIM + d];
        float inv = __expf(-ls);
        float iv2 = inv * inv;
        cA[m * D_DIM + d] = -0.5f * iv2;
        cB[m * D_DIM + d] = mu * iv2;
        cC[m * D_DIM + d] = fmaf(-0.5f * mu * mu, iv2, -ls) - 0.5f * LOG2PI - u;
    }
}

// ---------------------------------------------------------------------------
// P2: lsm[o,d] = logsumexp over m of weights_per_dim[o,m,d]
__global__ void prep_lsm(const float* __restrict__ wpd, float* __restrict__ lsm) {
    int t = blockIdx.x * blockDim.x + threadIdx.x;
    if (t >= O_DIM * D_DIM) return;
    int o = t / D_DIM, d = t % D_DIM;
    float v[M_DIM];
    float mx = -INFINITY;
    for (int m = 0; m < M_DIM; ++m) {
        v[m] = wpd[(o * M_DIM + m) * D_DIM + d];
        mx = fmaxf(mx, v[m]);
    }
    float s = 0.f;
    for (int m = 0; m < M_DIM; ++m) s += __expf(v[m] - mx);
    lsm[t] = mx + __logf(s);
}

// ---------------------------------------------------------------------------
// P3: wexp swizzled into the f32 WMMA B-matrix register layout, stored
// [d][lane(32)][i(16)] so each d-row is a contiguous 2KB block (one async
// LDS tile). B matrix per d: K = m (32), N = o (16).
__global__ void prep_wexp(const float* __restrict__ wpd,
                          const float* __restrict__ lsm,
                          float* __restrict__ wsw) {
    int t = blockIdx.x * blockDim.x + threadIdx.x;
    if (t >= D_DIM * 32 * 16) return;
    int i = t & 15;
    int lane = (t >> 4) & 31;
    int d = t >> 9;
    int m = k_of(i, lane);
    int o = lane & 15;
    wsw[t] = __expf(wpd[(o * M_DIM + m) * D_DIM + d] - lsm[o * D_DIM + d]);
}

// ---------------------------------------------------------------------------
// P4: class-weight exp(log_softmax) swizzled into B layout for stage 3.
// Layout [ctile(7)][i(8)][lane(32)]; K = o (16), N = c; padded c -> 0.
__global__ void prep_wexp2(const float* __restrict__ cw, float* __restrict__ w2) {
    int t = blockIdx.x * blockDim.x + threadIdx.x;
    if (t >= NCT * 8 * 32) return;
    int lane = t & 31;
    int i = (t >> 5) & 7;
    int ct = t >> 8;
    int c = ct * 16 + (lane & 15);
    int o = k_of(i, lane);
    float val = 0.f;
    if (c < C_DIM) {
        float mx = -INFINITY;
        for (int oo = 0; oo < O_DIM; ++oo) mx = fmaxf(mx, cw[c * O_DIM + oo]);
        float s = 0.f;
        for (int oo = 0; oo < O_DIM; ++oo) s += __expf(cw[c * O_DIM + oo] - mx);
        float lse = mx + __logf(s);
        val = __expf(cw[c * O_DIM + o] - lse);
    }
    w2[t] = val;
}

// ---------------------------------------------------------------------------
// Main kernel: one wave per (b-tile of 16, d-group of 32 d's).
// Per d: build A = exp(base - U[d]) in WMMA A-layout (2 FMA + 1 exp per elem),
// stream B (wexp) into LDS with double-buffered global_load_async_to_lds_b128
// (ASYNCcnt-tracked), run 8 chained V_WMMA_F32_16X16X4_F32 to get
// C[b,o] = sum_m exp(base-U)*wexp, then accumulate log(C) over d.
__global__ void __launch_bounds__(32)
main_kernel(const float* __restrict__ z, const float* __restrict__ cA,
            const float* __restrict__ cB, const float* __restrict__ cC,
            const float* __restrict__ wsw, float* __restrict__ rpart) {
    __shared__ float lbuf[2][512];            // 2 x 2KB double buffer

    const int lane = threadIdx.x;
    const int bt = blockIdx.x, dg = blockIdx.y;
    const int brow = lane & 15;
    const int mofs = (lane & 16) ? 2 : 0;
    const int d0 = dg * ND;

    float accL[8];
#pragma unroll
    for (int v = 0; v < 8; ++v) accL[v] = 0.f;

    const float* zrow = z + (bt * 16 + brow) * D_DIM;
    __builtin_prefetch(zrow + d0, 0, 1);      // global_prefetch_b8

    // Per-lane LDS byte offset of this lane's 64B slice of each buffer
    // (flat shared address truncates to the wave-relative LDS offset).
    const unsigned lds0 = (unsigned)(uintptr_t)(&lbuf[0][0]) + (unsigned)lane * 64u;

    // Issue async fill of buffer 0 with the d0 wexp row: 4 x 512B, each
    // instruction moves 16B/lane; INST_OFFSET applies to both LDS and global.
    {
        const float* g = wsw + (size_t)(d0 * 32 + lane) * 16;
        asm volatile(
            "global_load_async_to_lds_b128 %0, %1, off\n\t"
            "global_load_async_to_lds_b128 %0, %1, off offset:16\n\t"
            "global_load_async_to_lds_b128 %0, %1, off offset:32\n\t"
            "global_load_async_to_lds_b128 %0, %1, off offset:48"
            :: "v"(lds0), "v"(g) : "memory");
    }

    int buf = 0;
    for (int dd = 0; dd < ND; ++dd) {
        const int d = d0 + dd;
        const float zb = zrow[d];

        if (dd + 1 < ND) {
            // WAR guard: our ds reads of the buffer we are about to refill
            // must have drained before the async engine rewrites it.
            asm volatile("s_wait_dscnt 0x0" ::: "memory");
            const float* g = wsw + (size_t)((d + 1) * 32 + lane) * 16;
            const unsigned l = lds0 + (unsigned)((buf ^ 1) * 2048);
            asm volatile(
                "global_load_async_to_lds_b128 %0, %1, off\n\t"
                "global_load_async_to_lds_b128 %0, %1, off offset:16\n\t"
                "global_load_async_to_lds_b128 %0, %1, off offset:32\n\t"
                "global_load_async_to_lds_b128 %0, %1, off offset:48"
                :: "v"(l), "v"(g) : "memory");
            // current buffer's 4 loads done; only the 4 new ones outstanding
            asm volatile("s_wait_asynccnt 0x4" ::: "memory");
        } else {
            asm volatile("s_wait_asynccnt 0x0" ::: "memory");
        }

        // B operand from LDS: 16 contiguous floats per lane -> ds_load_b128
        const float* wl = &lbuf[buf][lane * 16];
        float w[16];
#pragma unroll
        for (int i = 0; i < 16; ++i) w[i] = wl[i];

        // A operand: e[i] = exp(A*z^2 + B*z + C') for this lane's (b, m) slots
        float e[16];
#pragma unroll
        for (int i = 0; i < 16; ++i) {
            int m = 4 * (i >> 1) + (i & 1) + mofs;
            float a = cA[m * D_DIM + d];
            float bb = cB[m * D_DIM + d];
            float cc = cC[m * D_DIM + d];
            e[i] = __expf(fmaf(fmaf(a, zb, bb), zb, cc));
        }

        v8f acc = {};
#pragma unroll
        for (int k = 0; k < 8; ++k) {
            v2f av; av.x = e[2 * k]; av.y = e[2 * k + 1];
            v2f bv; bv.x = w[2 * k]; bv.y = w[2 * k + 1];
            acc = __builtin_amdgcn_wmma_f32_16x16x4_f32(
                false, av, false, bv, (short)0, acc, false, false);
        }

        // logsumexp_m = U[d] + log(C); U folded in later as scalar S
#pragma unroll
        for (int v = 0; v < 8; ++v) accL[v] += fast_log(acc[v]);

        buf ^= 1;
    }

    // C/D layout: VGPR v holds row b = v + (lane>=16 ? 8 : 0), col o = lane%16
    const int rbase = ((dg * NBT + bt) * 16) * 16;
#pragma unroll
    for (int v = 0; v < 8; ++v) {
        int row = v + ((lane & 16) ? 8 : 0);
        rpart[rbase + row * 16 + (lane & 15)] = accL[v];
    }
}

// ---------------------------------------------------------------------------
// Stage 3: one wave per (b-tile, c-tile). r[b,o] = sum of DG partials; then
// out[b,c] = S + max_o r + log( exp(r - max) @ exp(log_softmax(cw)) )
// via 4 chained V_WMMA_F32_16X16X4_F32 (K = O = 16).
__global__ void __launch_bounds__(32)
stage3_kernel(const float* __restrict__ rpart, const float* __restrict__ w2,
              const float* __restrict__ U, float* __restrict__ out) {
    const int lane = threadIdx.x;
    const int bt = blockIdx.x, ct = blockIdx.y;
    const int brow = lane & 15;
    const int mofs = (lane & 16) ? 2 : 0;

    // Scalar S = sum_d U[d] (broadcast loads; deterministic)
    float S = 0.f;
    for (int i = 0; i < D_DIM; ++i) S += U[i];

    // Gather r in A layout: per lane 8 values, o in {4k+mofs, 4k+1+mofs}
    float r[8];
#pragma unroll
    for (int i = 0; i < 8; ++i) {
        int o = 4 * (i >> 1) + (i & 1) + mofs;
        float s = 0.f;
        for (int g = 0; g < DG; ++g)
            s += rpart[((g * NBT + bt) * 16 + brow) * 16 + o];
        r[i] = s;
    }

    float mx = r[0];
#pragma unroll
    for (int i = 1; i < 8; ++i) mx = fmaxf(mx, r[i]);
    mx = fmaxf(mx, __shfl_xor(mx, 16, 32));   // lane pair (l, l^16) shares b

    float a[8];
#pragma unroll
    for (int i = 0; i < 8; ++i) a[i] = __expf(r[i] - mx);

    const float* wl = w2 + ct * 8 * 32;
    v8f acc = {};
#pragma unroll
    for (int k = 0; k < 4; ++k) {
        v2f av; av.x = a[2 * k];                av.y = a[2 * k + 1];
        v2f bv; bv.x = wl[(2 * k) * 32 + lane]; bv.y = wl[(2 * k + 1) * 32 + lane];
        acc = __builtin_amdgcn_wmma_f32_16x16x4_f32(
            false, av, false, bv, (short)0, acc, false, false);
    }

#pragma unroll
    for (int v = 0; v < 8; ++v) {
        int row = v + ((lane & 16) ? 8 : 0);
        float mrow = __shfl(mx, row, 32);   // row max lives at lane `row`
        int c = ct * 16 + (lane & 15);
        if (c < C_DIM)
            out[(bt * 16 + row) * C_DIM + c] = S + mrow + fast_log(acc[v]);
    }
}

// ---------------------------------------------------------------------------
__global__ void copy_z(const float* __restrict__ z, float* __restrict__ out, int n) {
    int i = blockIdx.x * blockDim.x + threadIdx.x;
    if (i < n) out[i] = z[i];
}

// ---------------------------------------------------------------------------
extern "C" void kernel_launch(void* const* d_in, const int* in_sizes, int n_in,
                              void* d_out, int out_size, void* d_ws, size_t ws_size,
                              hipStream_t stream) {
    (void)in_sizes; (void)n_in; (void)out_size; (void)ws_size;

    const float* z   = (const float*)d_in[0];   // [B, D]
    const float* mu  = (const float*)d_in[1];   // [M, D]
    const float* ls  = (const float*)d_in[2];   // [M, D]
    const float* wpd = (const float*)d_in[3];   // [O, M, D]
    const float* cw  = (const float*)d_in[4];   // [C, O]

    float* out = (float*)d_out;                 // z (B*D) ++ result (B*C)

    // Workspace carve-up (floats): ~1.7 MB total
    float* ws    = (float*)d_ws;
    float* cA    = ws;                      // 32*256
    float* cB    = cA + M_DIM * D_DIM;      // 32*256
    float* cC    = cB + M_DIM * D_DIM;      // 32*256
    float* U     = cC + M_DIM * D_DIM;      // 256
    float* lsm   = U + D_DIM;               // 16*256
    float* wsw   = lsm + O_DIM * D_DIM;     // 256*32*16
    float* w2    = wsw + D_DIM * 32 * 16;   // 7*8*32
    float* rpart = w2 + NCT * 8 * 32;       // 8*128*16*16

    prep_coefs<<<1, 256, 0, stream>>>(mu, ls, cA, cB, cC, U);
    prep_lsm<<<(O_DIM * D_DIM + 127) / 128, 128, 0, stream>>>(wpd, lsm);
    prep_wexp<<<(D_DIM * 32 * 16) / 256, 256, 0, stream>>>(wpd, lsm, wsw);
    prep_wexp2<<<(NCT * 8 * 32 + 255) / 256, 256, 0, stream>>>(cw, w2);

    main_kernel<<<dim3(NBT, DG), 32, 0, stream>>>(z, cA, cB, cC, wsw, rpart);

    stage3_kernel<<<dim3(NBT, NCT), 32, 0, stream>>>(rpart, w2, U,
                                                     out + B_DIM * D_DIM);

    copy_z<<<(B_DIM * D_DIM + 255) / 256, 256, 0, stream>>>(z, out, B_DIM * D_DIM);
}